// CFGGNN_78477642432722
// MI455X (gfx1250) — compile-verified
//
#include <hip/hip_runtime.h>
#include <hip/hip_bf16.h>

// ---------------------------------------------------------------------------
// CFG-GNN forward: 3x GCNConv (fp32 WMMA GEMM + edge scatter) + mean pool + cls
// Target: gfx1250 (CDNA5, wave32, V_WMMA_F32_16X16X4_F32)
// GEMM is register-blocked: each wave computes 4 M-tiles (64 rows x 16 cols),
// so each B fragment feeds 4 v_wmma instructions.
// ---------------------------------------------------------------------------

typedef float v2f __attribute__((ext_vector_type(2)));
typedef float v8f __attribute__((ext_vector_type(8)));

#define NNODES 40000
#define NEDGES 640000
#define NGRAPH 64
#define DIM_IN 128
#define DIM_H  256
#define DIM_O  128
#define MTILES 4   // M-tiles per wave (64 rows)

// ---------------- elementwise helpers ----------------

__global__ void k_fill(float* __restrict__ p, float v, long n) {
    long i = (long)blockIdx.x * blockDim.x + threadIdx.x;
    if (i < n) p[i] = v;
}

__global__ void k_deg_accum(const int* __restrict__ dst, float* __restrict__ deg, int e) {
    int i = blockIdx.x * blockDim.x + threadIdx.x;
    if (i < e) unsafeAtomicAdd(&deg[dst[i]], 1.0f);
}

__global__ void k_rsqrt(float* __restrict__ d, int n) {
    int i = blockIdx.x * blockDim.x + threadIdx.x;
    if (i < n) d[i] = rsqrtf(d[i]);
}

// ---------------- fp32 WMMA GEMM: C[M x Nc] = A[M x K] @ W[K x Nc] ----------
// Block = 256 threads = 8 waves.  Wave w owns col-tile (blockIdx.y*8+w) and
// MTILES row-tiles starting at blockIdx.x*16*MTILES.  K-loop steps by 4.
// grid = (M/(16*MTILES), Nc/128)

__global__ __launch_bounds__(256) void k_gemm_wmma(const float* __restrict__ A,
                                                   const float* __restrict__ W,
                                                   float* __restrict__ C,
                                                   int M, int K, int Nc) {
    const int lane = threadIdx.x & 31;
    const int wave = threadIdx.x >> 5;
    const int m0 = blockIdx.x * (16 * MTILES);
    const int n0 = (blockIdx.y * 8 + wave) * 16;
    if (m0 >= M || n0 >= Nc) return;

    const bool hi = lane >= 16;
    const int  l  = lane & 15;
    const int  col = n0 + l;

    const float* __restrict__ arow[MTILES];
#pragma unroll
    for (int t = 0; t < MTILES; ++t)
        arow[t] = A + (size_t)(m0 + t * 16 + l) * K;

    v8f acc[MTILES];
#pragma unroll
    for (int t = 0; t < MTILES; ++t) acc[t] = (v8f){};

    for (int k0 = 0; k0 < K; k0 += 4) {
        const int ka = k0 + (hi ? 2 : 0);
        v2f b;
        b.x = W[(size_t)ka * Nc + col];
        b.y = W[(size_t)(ka + 1) * Nc + col];
#pragma unroll
        for (int t = 0; t < MTILES; ++t) {
            v2f a = *(const v2f*)(arow[t] + ka);
            acc[t] = __builtin_amdgcn_wmma_f32_16x16x4_f32(
                /*neg_a=*/false, a, /*neg_b=*/false, b,
                /*c_mod=*/(short)0, acc[t], /*reuse_a=*/false, /*reuse_b=*/false);
        }
    }

#pragma unroll
    for (int t = 0; t < MTILES; ++t) {
        const int rbase = m0 + t * 16 + (hi ? 8 : 0);
#pragma unroll
        for (int v = 0; v < 8; ++v)
            C[(size_t)(rbase + v) * Nc + col] = acc[t][v];
    }
}

// ---------------- edge scatter: agg[dst] += dis[src]*dis[dst]*h[src] --------
// H/4 threads per edge (power of 2), each does a float4 gather + 4 atomics.

__global__ void k_scatter(const float* __restrict__ h, const int* __restrict__ src,
                          const int* __restrict__ dst, const float* __restrict__ dis,
                          float* __restrict__ agg, long total, int shift, int H) {
    long gid = (long)blockIdx.x * blockDim.x + threadIdx.x;
    if (gid >= total) return;
    const int e = (int)(gid >> shift);
    const int j = (int)((gid & ((1 << shift) - 1)) << 2);
    const int s = src[e];
    const int d = dst[e];
    const float nrm = dis[s] * dis[d];
    const float4 hv = *(const float4*)(h + (size_t)s * H + j);
    float* out = agg + (size_t)d * H + j;
    unsafeAtomicAdd(out + 0, nrm * hv.x);
    unsafeAtomicAdd(out + 1, nrm * hv.y);
    unsafeAtomicAdd(out + 2, nrm * hv.z);
    unsafeAtomicAdd(out + 3, nrm * hv.w);
}

// ---------------- fuse: out = [relu](agg + dis^2 * h + b) -------------------

__global__ void k_fuse(const float* __restrict__ agg, const float* __restrict__ h,
                       const float* __restrict__ dis, const float* __restrict__ b,
                       float* __restrict__ out, long total, int H, int relu) {
    long gid = (long)blockIdx.x * blockDim.x + threadIdx.x;
    if (gid >= total) return;
    const int i = (int)(gid / H);
    const int j = (int)(gid - (long)i * H);
    const float ds = dis[i];
    float v = agg[gid] + ds * ds * h[gid] + b[j];
    if (relu) v = fmaxf(v, 0.0f);
    out[gid] = v;
}

// ---------------- pooling ---------------------------------------------------

__global__ void k_count(const int* __restrict__ batch, float* __restrict__ counts, int n) {
    int i = blockIdx.x * blockDim.x + threadIdx.x;
    if (i < n) unsafeAtomicAdd(&counts[batch[i]], 1.0f);
}

__global__ void k_pool_sum(const float* __restrict__ emb, const int* __restrict__ batch,
                           float* __restrict__ pool, long total, int D) {
    long gid = (long)blockIdx.x * blockDim.x + threadIdx.x;
    if (gid >= total) return;
    const int i = (int)(gid / D);
    const int j = (int)(gid - (long)i * D);
    unsafeAtomicAdd(&pool[(size_t)batch[i] * D + j], emb[gid]);
}

__global__ void k_pool_fin(const float* __restrict__ pool, const float* __restrict__ counts,
                           float* __restrict__ gemb, int total, int D) {
    int gid = blockIdx.x * blockDim.x + threadIdx.x;
    if (gid >= total) return;
    const int g = gid / D;
    gemb[gid] = pool[gid] / fmaxf(counts[g], 1.0f);
}

// ---------------- classifier: logits[G x 2] = gemb[G x D] @ Wcls[D x 2] + b --

__global__ void k_logits(const float* __restrict__ gemb, const float* __restrict__ Wc,
                         const float* __restrict__ bc, float* __restrict__ logits, int D) {
    const int t = threadIdx.x;            // 128 threads: (g, c) pairs
    const int g = t >> 1;
    const int c = t & 1;
    float acc = bc[c];
    const float* row = gemb + (size_t)g * D;
    for (int k = 0; k < D; ++k) acc += row[k] * Wc[k * 2 + c];
    logits[g * 2 + c] = acc;
}

// ---------------------------------------------------------------------------

static inline int cdiv_l(long a, int b) { return (int)((a + b - 1) / b); }

extern "C" void kernel_launch(void* const* d_in, const int* in_sizes, int n_in,
                              void* d_out, int out_size, void* d_ws, size_t ws_size,
                              hipStream_t stream) {
    const float* x     = (const float*)d_in[0];
    const int*   ei    = (const int*)d_in[1];
    const int*   batch = (const int*)d_in[2];
    const float* W_in  = (const float*)d_in[3];
    const float* b_in  = (const float*)d_in[4];
    const float* W_mid = (const float*)d_in[5];
    const float* b_mid = (const float*)d_in[6];
    const float* W_out = (const float*)d_in[7];
    const float* b_out = (const float*)d_in[8];
    const float* W_cls = (const float*)d_in[9];
    const float* b_cls = (const float*)d_in[10];

    const int N = in_sizes[0] / DIM_IN;   // 40000
    const int E = in_sizes[1] / 2;        // 640000
    const int* src = ei;
    const int* dst = ei + E;

    // workspace layout (floats)
    float* w       = (float*)d_ws;
    float* dis     = w;                            // N
    float* bufH    = dis + N;                      // N*256  (GEMM output)
    float* bufAgg  = bufH + (size_t)N * DIM_H;     // N*256  (scatter accumulator)
    float* bufA    = bufAgg + (size_t)N * DIM_H;   // N*256  (activated layer out)
    float* pool    = bufA + (size_t)N * DIM_H;     // G*128
    float* counts  = pool + (size_t)NGRAPH * DIM_O;// G

    float* node_emb = (float*)d_out;                       // N*128
    float* gemb     = node_emb + (size_t)N * DIM_O;        // G*128
    float* logits   = gemb + (size_t)NGRAPH * DIM_O;       // G*2

    const int B = 256;
    const int MB = 16 * MTILES;           // rows per block

    // ---- degree (with self loop) -> dis = rsqrt(deg) ----
    k_fill<<<cdiv_l(N, B), B, 0, stream>>>(dis, 1.0f, N);
    k_deg_accum<<<cdiv_l(E, B), B, 0, stream>>>(dst, dis, E);
    k_rsqrt<<<cdiv_l(N, B), B, 0, stream>>>(dis, N);

    // ---- layer 1: h = x @ W_in ; propagate ; relu ----
    {
        dim3 grid(N / MB, DIM_H / 128);
        k_gemm_wmma<<<grid, B, 0, stream>>>(x, W_in, bufH, N, DIM_IN, DIM_H);
        long tot = (long)N * DIM_H;
        k_fill<<<cdiv_l(tot, B), B, 0, stream>>>(bufAgg, 0.0f, tot);
        long etot = (long)E * (DIM_H / 4);
        k_scatter<<<cdiv_l(etot, B), B, 0, stream>>>(bufH, src, dst, dis, bufAgg, etot, 6, DIM_H);
        k_fuse<<<cdiv_l(tot, B), B, 0, stream>>>(bufAgg, bufH, dis, b_in, bufA, tot, DIM_H, 1);
    }

    // ---- layer 2: h = bufA @ W_mid ; propagate ; relu ----
    {
        dim3 grid(N / MB, DIM_H / 128);
        k_gemm_wmma<<<grid, B, 0, stream>>>(bufA, W_mid, bufH, N, DIM_H, DIM_H);
        long tot = (long)N * DIM_H;
        k_fill<<<cdiv_l(tot, B), B, 0, stream>>>(bufAgg, 0.0f, tot);
        long etot = (long)E * (DIM_H / 4);
        k_scatter<<<cdiv_l(etot, B), B, 0, stream>>>(bufH, src, dst, dis, bufAgg, etot, 6, DIM_H);
        k_fuse<<<cdiv_l(tot, B), B, 0, stream>>>(bufAgg, bufH, dis, b_mid, bufA, tot, DIM_H, 1);
    }

    // ---- layer 3: node_emb = bufA @ W_out ; propagate (no relu) ----
    {
        dim3 grid(N / MB, DIM_O / 128);
        k_gemm_wmma<<<grid, B, 0, stream>>>(bufA, W_out, bufH, N, DIM_H, DIM_O);
        long tot = (long)N * DIM_O;
        k_fill<<<cdiv_l(tot, B), B, 0, stream>>>(bufAgg, 0.0f, tot);
        long etot = (long)E * (DIM_O / 4);
        k_scatter<<<cdiv_l(etot, B), B, 0, stream>>>(bufH, src, dst, dis, bufAgg, etot, 5, DIM_O);
        k_fuse<<<cdiv_l(tot, B), B, 0, stream>>>(bufAgg, bufH, dis, b_out, node_emb, tot, DIM_O, 0);
    }

    // ---- global mean pool ----
    {
        long ptot = (long)NGRAPH * DIM_O + NGRAPH;  // pool + counts contiguous
        k_fill<<<cdiv_l(ptot, B), B, 0, stream>>>(pool, 0.0f, ptot);
        k_count<<<cdiv_l(N, B), B, 0, stream>>>(batch, counts, N);
        long tot = (long)N * DIM_O;
        k_pool_sum<<<cdiv_l(tot, B), B, 0, stream>>>(node_emb, batch, pool, tot, DIM_O);
        int gtot = NGRAPH * DIM_O;
        k_pool_fin<<<cdiv_l(gtot, B), B, 0, stream>>>(pool, counts, gemb, gtot, DIM_O);
    }

    // ---- classifier ----
    k_logits<<<1, 128, 0, stream>>>(gemb, W_cls, b_cls, logits, DIM_O);
}